// Model_40810779247488
// MI455X (gfx1250) — compile-verified
//
#include <hip/hip_runtime.h>
#include <hip/hip_bf16.h>
#include <stdint.h>

#define B_   2048
#define E_   20
#define K_   5
#define P_   336
#define C_   21
#define N_   (B_ * K_)          // 10240 nonzeros
#define RC_  (P_ * C_)          // 7056 floats per row
#define RV_  (RC_ / 4)          // 1764 float4 per row
#define TPB  256
#define NT_  ((RV_ + TPB - 1) / TPB)   // 7 tiles
#define EPS_F 2.220446049250313e-16f

// ---------------------------------------------------------------------------
// Kernel 1: deterministic inverse index, no atomics.
// Nonzeros are expert-major, batch-ascending within expert, so the rank of
// row i among batch b's K rows = #selected experts of b with id < e.
// ---------------------------------------------------------------------------
__global__ __launch_bounds__(256) void build_inverse_kernel(
    const float* __restrict__ gates, const int* __restrict__ bidx,
    const int* __restrict__ eidx, int* __restrict__ inv_row,
    float* __restrict__ inv_gate) {
  int i = blockIdx.x * blockDim.x + threadIdx.x;
  if (i >= N_) return;
  int b = bidx[i];
  int e = eidx[i];
  const float* g = gates + b * E_;
  float gv = g[e];
  int rank = 0;
  for (int e2 = 0; e2 < e; ++e2) rank += (g[e2] != 0.0f) ? 1 : 0;
  inv_row[b * K_ + rank]  = i;
  inv_gate[b * K_ + rank] = gv;
}

// CDNA5 async copy: per-lane 16B global -> LDS, tracked by ASYNCcnt.
__device__ __forceinline__ void async_b128(uint32_t lds_off, const float4* gaddr) {
  asm volatile("global_load_async_to_lds_b128 %0, %1, off"
               :: "v"(lds_off), "v"(gaddr)
               : "memory");
}

// ---------------------------------------------------------------------------
// Kernel 2: one block per output row b. Double-buffered async-LDS pipeline:
// stage 5 gathered row-tiles for tile t+1 while computing tile t.
// Each lane writes/reads its own LDS slot -> no intra-block barrier needed.
// Transcendentals use the hardware TRANS pipe (v_exp_f32 / v_log_f32) so the
// kernel stays HBM-bound instead of VALU-bound.
// ---------------------------------------------------------------------------
__global__ __launch_bounds__(TPB) void combine_kernel(
    const float4* __restrict__ xs, const int* __restrict__ inv_row,
    const float* __restrict__ inv_gate, float4* __restrict__ out) {
  __shared__ float4 lbuf[2][K_][TPB];          // 40 KB -> 8 blocks / 320KB WGP
  __shared__ int    s_row[K_];
  __shared__ float  s_gate[K_];

  const int b   = blockIdx.x;
  const int tid = threadIdx.x;
  if (tid < K_) {
    s_row[tid]  = inv_row[b * K_ + tid];
    s_gate[tid] = inv_gate[b * K_ + tid];
  }
  __syncthreads();

  const float4* src[K_];
  float g[K_];
#pragma unroll
  for (int k = 0; k < K_; ++k) {
    src[k] = xs + (size_t)s_row[k] * RV_;
    g[k]   = s_gate[k];
  }
  float4* dst = out + (size_t)b * RV_;

  // Prologue: stage tile 0 into buffer 0 (tid < 256 <= RV_, always in range).
#pragma unroll
  for (int k = 0; k < K_; ++k)
    async_b128((uint32_t)(uintptr_t)&lbuf[0][k][tid], src[k] + tid);

  for (int t = 0; t < NT_; ++t) {
    const int cur = t & 1;
    if (t + 1 < NT_) {
      // Stage next tile; clamp index so every lane issues (keeps ASYNCcnt
      // bookkeeping exact; duplicate loads on the tail are harmless).
      int nidx = (t + 1) * TPB + tid;
      if (nidx > RV_ - 1) nidx = RV_ - 1;
#pragma unroll
      for (int k = 0; k < K_; ++k)
        async_b128((uint32_t)(uintptr_t)&lbuf[cur ^ 1][k][tid], src[k] + nidx);
      // 5 loads of tile t + 5 of tile t+1 may be in flight; async loads
      // complete in order, so <=5 means tile t's data has landed in LDS.
      asm volatile("s_wait_asynccnt 0x5" ::: "memory");
    } else {
      asm volatile("s_wait_asynccnt 0x0" ::: "memory");
    }

    const int idx = t * TPB + tid;
    if (idx < RV_) {
      float4 acc;
      acc.x = acc.y = acc.z = acc.w = 0.0f;
#pragma unroll
      for (int k = 0; k < K_; ++k) {
        float4 v = lbuf[cur][k][tid];
        float gk = g[k];
        acc.x = fmaf(gk, __expf(v.x), acc.x);
        acc.y = fmaf(gk, __expf(v.y), acc.y);
        acc.z = fmaf(gk, __expf(v.z), acc.z);
        acc.w = fmaf(gk, __expf(v.w), acc.w);
      }
      acc.x = __logf(acc.x == 0.0f ? EPS_F : acc.x);
      acc.y = __logf(acc.y == 0.0f ? EPS_F : acc.y);
      acc.z = __logf(acc.z == 0.0f ? EPS_F : acc.z);
      acc.w = __logf(acc.w == 0.0f ? EPS_F : acc.w);
      dst[idx] = acc;
    }
  }
}

extern "C" void kernel_launch(void* const* d_in, const int* in_sizes, int n_in,
                              void* d_out, int out_size, void* d_ws, size_t ws_size,
                              hipStream_t stream) {
  const float* xs    = (const float*)d_in[0];   // [N, P, C] f32
  const float* gates = (const float*)d_in[1];   // [B, E]    f32
  const int*   bidx  = (const int*)d_in[2];     // [N]       i32
  const int*   eidx  = (const int*)d_in[3];     // [N]       i32
  float*       out   = (float*)d_out;           // [B, P, C] f32

  int*   inv_row  = (int*)d_ws;                               // N_ ints
  float* inv_gate = (float*)((char*)d_ws + N_ * sizeof(int)); // N_ floats

  build_inverse_kernel<<<(N_ + 255) / 256, 256, 0, stream>>>(
      gates, bidx, eidx, inv_row, inv_gate);

  combine_kernel<<<B_, TPB, 0, stream>>>(
      (const float4*)xs, inv_row, inv_gate, (float4*)out);
}